// DrugEncoder_49057116455563
// MI455X (gfx1250) — compile-verified
//
#include <hip/hip_runtime.h>
#include <cstddef>
#include <cstdint>

typedef float v2f __attribute__((ext_vector_type(2)));
typedef float v8f __attribute__((ext_vector_type(8)));

#define GEMM_BLOCK 256

// ---------------------------------------------------------------------------
// FAST f32 WMMA GEMM: C[M,N] = A[M,K] @ W[K,N] + bias (+relu)
// Requirements: M%16==0, N%32==0, K%64==0, lda%4==0 (16B-aligned rows).
// Block = 8 waves; each wave computes a 16x32 tile with V_WMMA_F32_16X16X4_F32.
// A tile (16x64) staged via GLOBAL_LOAD_ASYNC_TO_LDS_B128 (CDNA5 async path).
// B addressing strength-reduced to pointer increments (2 adds per k-step).
// ---------------------------------------------------------------------------
__global__ __launch_bounds__(GEMM_BLOCK)
void gemm_wmma_f32_fast(const float* __restrict__ A, int lda,
                        const float* __restrict__ W, int ldw,
                        const float* __restrict__ bias,
                        float* __restrict__ C, int ldc,
                        int M, int K, int N, int relu)
{
  __shared__ float lds_a[16 * 64];
  const int tid  = threadIdx.x;
  const int wave = tid >> 5;
  const int lane = tid & 31;
  const int half = lane >> 4;
  const int ln   = lane & 15;

  const int m0 = blockIdx.y * 16;
  const int n0 = blockIdx.x * 256 + wave * 32;
  const bool active = (n0 < N);   // wave-uniform

  // Each thread owns 4 consecutive floats of the 16x64 A tile (one B128).
  const int ti = tid * 4;
  const int tr = ti >> 6;          // row 0..15
  const int tc = ti & 63;          // col 0..60 step 4
  const unsigned lds_dst = (unsigned)(uintptr_t)(lds_a + ti);  // LDS byte offset

  v8f acc0 = {0.f,0.f,0.f,0.f,0.f,0.f,0.f,0.f};
  v8f acc1 = {0.f,0.f,0.f,0.f,0.f,0.f,0.f,0.f};

  // Per-lane B row pointers: rows (2*half) and (2*half+1), column n0+ln.
  const float* wp0 = W + (size_t)(2 * half) * ldw + n0 + ln;
  const float* wp1 = wp0 + ldw;
  const size_t wstep = (size_t)4 * ldw;   // advance 4 K-rows per step

  for (int k0 = 0; k0 < K; k0 += 64) {
    // Async DMA: global -> LDS, 16 bytes per lane, no VGPR round trip.
    {
      const float* gp = A + (size_t)(m0 + tr) * lda + (k0 + tc);
      asm volatile("global_load_async_to_lds_b128 %0, %1, off"
                   :: "v"(lds_dst), "v"(gp) : "memory");
      asm volatile("s_wait_asynccnt 0" ::: "memory");
    }
    __syncthreads();
    if (active) {
      const int abase = ln * 64 + 2 * half;
#pragma unroll
      for (int kk = 0; kk < 64; kk += 4) {
        v2f a;
        a.x = lds_a[abase + kk];
        a.y = lds_a[abase + kk + 1];
        v2f b0, b1;
        b0.x = wp0[0];
        b0.y = wp1[0];
        b1.x = wp0[16];
        b1.y = wp1[16];
        acc0 = __builtin_amdgcn_wmma_f32_16x16x4_f32(false, a, false, b0,
                                                     (short)0, acc0, false, false);
        acc1 = __builtin_amdgcn_wmma_f32_16x16x4_f32(false, a, false, b1,
                                                     (short)0, acc1, false, false);
        wp0 += wstep;
        wp1 += wstep;
      }
    } else {
      wp0 += (size_t)16 * wstep;   // keep pointers consistent (not used)
      wp1 += (size_t)16 * wstep;
    }
    __syncthreads();
  }

  if (active) {
    const float bb0 = bias ? bias[n0 + ln]      : 0.f;
    const float bb1 = bias ? bias[n0 + 16 + ln] : 0.f;
#pragma unroll
    for (int j = 0; j < 8; j++) {
      const int row = m0 + j + 8 * half;
      float o0 = acc0[j] + bb0;
      float o1 = acc1[j] + bb1;
      if (relu) { o0 = fmaxf(o0, 0.f); o1 = fmaxf(o1, 0.f); }
      C[(size_t)row * ldc + n0 + ln]      = o0;
      C[(size_t)row * ldc + n0 + 16 + ln] = o1;
    }
  }
}

// ---------------------------------------------------------------------------
// GUARDED f32 WMMA GEMM for ragged K (e.g. K=9). Clamped (branchless) loads.
// Requirements: M%16==0, N%32==0.
// ---------------------------------------------------------------------------
__global__ __launch_bounds__(GEMM_BLOCK)
void gemm_wmma_f32_guard(const float* __restrict__ A, int lda,
                         const float* __restrict__ W, int ldw,
                         const float* __restrict__ bias,
                         float* __restrict__ C, int ldc,
                         int M, int K, int N, int relu)
{
  __shared__ float lds_a[16 * 32];
  const int tid  = threadIdx.x;
  const int wave = tid >> 5;
  const int lane = tid & 31;
  const int half = lane >> 4;
  const int ln   = lane & 15;

  const int m0 = blockIdx.y * 16;
  const int n0 = blockIdx.x * 256 + wave * 32;
  const bool active = (n0 < N);

  v8f acc0 = {0.f,0.f,0.f,0.f,0.f,0.f,0.f,0.f};
  v8f acc1 = {0.f,0.f,0.f,0.f,0.f,0.f,0.f,0.f};

  for (int k0 = 0; k0 < K; k0 += 32) {
#pragma unroll
    for (int it = 0; it < 2; it++) {           // 512 elements, 256 threads
      const int i  = tid + it * GEMM_BLOCK;
      const int r  = i >> 5;
      const int c  = i & 31;
      const int gk = k0 + c;
      const int gkc = gk < K ? gk : K - 1;     // clamp -> unconditional load
      const float v = A[(size_t)(m0 + r) * lda + gkc];
      lds_a[i] = gk < K ? v : 0.0f;
    }
    __syncthreads();
    if (active) {
#pragma unroll
      for (int kk = 0; kk < 32; kk += 4) {
        const int kb = kk + 2 * half;
        v2f a;
        a.x = lds_a[ln * 32 + kb];
        a.y = lds_a[ln * 32 + kb + 1];
        const int gk0 = k0 + kb;
        const int r0 = gk0     < K ? gk0     : K - 1;
        const int r1 = gk0 + 1 < K ? gk0 + 1 : K - 1;
        const float* w0 = W + (size_t)r0 * ldw + n0;
        const float* w1 = W + (size_t)r1 * ldw + n0;
        const bool k0ok = (gk0 < K), k1ok = (gk0 + 1 < K);
        v2f b0, b1;
        b0.x = k0ok ? w0[ln]      : 0.f;
        b0.y = k1ok ? w1[ln]      : 0.f;
        b1.x = k0ok ? w0[16 + ln] : 0.f;
        b1.y = k1ok ? w1[16 + ln] : 0.f;
        acc0 = __builtin_amdgcn_wmma_f32_16x16x4_f32(false, a, false, b0,
                                                     (short)0, acc0, false, false);
        acc1 = __builtin_amdgcn_wmma_f32_16x16x4_f32(false, a, false, b1,
                                                     (short)0, acc1, false, false);
      }
    }
    __syncthreads();
  }

  if (active) {
    const float bb0 = bias ? bias[n0 + ln]      : 0.f;
    const float bb1 = bias ? bias[n0 + 16 + ln] : 0.f;
#pragma unroll
    for (int j = 0; j < 8; j++) {
      const int row = m0 + j + 8 * half;
      float o0 = acc0[j] + bb0;
      float o1 = acc1[j] + bb1;
      if (relu) { o0 = fmaxf(o0, 0.f); o1 = fmaxf(o1, 0.f); }
      C[(size_t)row * ldc + n0 + ln]      = o0;
      C[(size_t)row * ldc + n0 + 16 + ln] = o1;
    }
  }
}

// ---------------------------------------------------------------------------
// Edge attention helpers
// ---------------------------------------------------------------------------
__device__ __forceinline__ unsigned f32_ord_enc(float f) {
  unsigned u = __float_as_uint(f);
  return (u & 0x80000000u) ? ~u : (u | 0x80000000u);
}
__device__ __forceinline__ float f32_ord_dec(unsigned e) {
  if (e == 0u) return 0.f;  // untouched (node with no in-edges)
  unsigned u = (e & 0x80000000u) ? (e & 0x7fffffffu) : ~e;
  return __uint_as_float(u);
}

// Pass A: one wave per (edge, head). alpha = q[dst]·(k[src]+e) * scale,
// segment-max via ordered-int atomicMax. Edge embedding computed on the fly.
__global__ __launch_bounds__(256)
void edge_alpha_max(const int* __restrict__ src, const int* __restrict__ dst,
                    const float* __restrict__ ea, const float* __restrict__ We,
                    const float* __restrict__ q, const float* __restrict__ k,
                    float* __restrict__ alpha, unsigned* __restrict__ amax,
                    int E, int H, int C, float scale)
{
  const int wid  = (int)(((unsigned)blockIdx.x * blockDim.x + threadIdx.x) >> 5);
  const int lane = threadIdx.x & 31;
  if (wid >= E * H) return;            // wave-uniform
  const int e = wid / H, h = wid - e * H;
  const int s = src[e], d = dst[e];
  const int HC = H * C;
  const float ea0 = ea[e*4+0], ea1 = ea[e*4+1], ea2 = ea[e*4+2], ea3 = ea[e*4+3];
  const float* qrow = q + (size_t)d * HC + h * C;
  const float* krow = k + (size_t)s * HC + h * C;
  const float* w    = We + h * C;      // row j of We at stride HC
  float part = 0.f;
  for (int c = lane; c < C; c += 32) {
    const float ev = ea0 * w[c] + ea1 * w[HC + c] + ea2 * w[2*HC + c] + ea3 * w[3*HC + c];
    part += qrow[c] * (krow[c] + ev);
  }
  for (int off = 16; off > 0; off >>= 1) part += __shfl_down(part, off, 32);
  if (lane == 0) {
    const float a = part * scale;
    alpha[(size_t)e * H + h] = a;
    atomicMax(&amax[(size_t)d * H + h], f32_ord_enc(a));
  }
}

// Pass B: ex = exp(alpha - max); scatter-add ex and ex*(v[src]+e).
__global__ __launch_bounds__(256)
void edge_scatter(const int* __restrict__ src, const int* __restrict__ dst,
                  const float* __restrict__ ea, const float* __restrict__ We,
                  const float* __restrict__ v, const float* __restrict__ alpha,
                  const unsigned* __restrict__ amax,
                  float* __restrict__ den, float* __restrict__ acc,
                  int E, int H, int C)
{
  const int wid  = (int)(((unsigned)blockIdx.x * blockDim.x + threadIdx.x) >> 5);
  const int lane = threadIdx.x & 31;
  if (wid >= E * H) return;            // wave-uniform
  const int e = wid / H, h = wid - e * H;
  const int s = src[e], d = dst[e];
  const int HC = H * C;
  const float a  = alpha[(size_t)e * H + h];
  const float mx = f32_ord_dec(amax[(size_t)d * H + h]);
  const float ex = __expf(a - mx);
  if (lane == 0) atomicAdd(&den[(size_t)d * H + h], ex);
  const float ea0 = ea[e*4+0], ea1 = ea[e*4+1], ea2 = ea[e*4+2], ea3 = ea[e*4+3];
  const float* vrow = v + (size_t)s * HC + h * C;
  float* arow = acc + (size_t)d * HC + h * C;
  const float* w = We + h * C;
  for (int c = lane; c < C; c += 32) {
    const float ev = ea0 * w[c] + ea1 * w[HC + c] + ea2 * w[2*HC + c] + ea3 * w[3*HC + c];
    atomicAdd(&arow[c], ex * (vrow[c] + ev));
  }
}

// Pass C: out = acc / (den + 1e-16) + skip (optional relu)
__global__ __launch_bounds__(256)
void node_finalize(const float* __restrict__ acc, const float* __restrict__ den,
                   const float* __restrict__ skip, float* __restrict__ out,
                   long total, int HC, int C, int H, int relu)
{
  const long idx = (long)blockIdx.x * 256 + threadIdx.x;
  if (idx >= total) return;
  const int n = (int)(idx / HC);
  const int j = (int)(idx - (long)n * HC);
  const int h = j / C;
  const float dsum = den[(size_t)n * H + h] + 1e-16f;
  float o = acc[idx] / dsum + skip[idx];
  out[idx] = relu ? fmaxf(o, 0.f) : o;
}

// global_mean_pool into the concat buffer (cols 0..HC-1 of row length ldg)
__global__ __launch_bounds__(256)
void pool_mean(const float* __restrict__ h, float* __restrict__ gcat,
               int Bg, int npg, int HC, int ldg)
{
  const int idx = blockIdx.x * 256 + threadIdx.x;
  if (idx >= Bg * HC) return;
  const int b = idx / HC, j = idx - b * HC;
  const float* p = h + (size_t)b * npg * HC + j;
  float s = 0.f;
  for (int t = 0; t < npg; t++) s += p[(size_t)t * HC];
  gcat[(size_t)b * ldg + j] = s * (1.0f / (float)npg);
}

// ---------------------------------------------------------------------------
extern "C" void kernel_launch(void* const* d_in, const int* in_sizes, int n_in,
                              void* d_out, int out_size, void* d_ws, size_t ws_size,
                              hipStream_t stream)
{
  (void)in_sizes; (void)n_in; (void)out_size; (void)ws_size;
  const int N = 24000, E = 96000, Bg = 1200, NPG = 20;

  const float* x    = (const float*)d_in[0];
  const int*   eidx = (const int*)  d_in[1];
  const float* ea   = (const float*)d_in[2];
  const float* fp   = (const float*)d_in[4];
  const float* Wq1 = (const float*)d_in[5],  *bq1 = (const float*)d_in[6];
  const float* Wk1 = (const float*)d_in[7],  *bk1 = (const float*)d_in[8];
  const float* Wv1 = (const float*)d_in[9],  *bv1 = (const float*)d_in[10];
  const float* We1 = (const float*)d_in[11];
  const float* Ws1 = (const float*)d_in[12], *bs1 = (const float*)d_in[13];
  const float* Wq2 = (const float*)d_in[14], *bq2 = (const float*)d_in[15];
  const float* Wk2 = (const float*)d_in[16], *bk2 = (const float*)d_in[17];
  const float* Wv2 = (const float*)d_in[18], *bv2 = (const float*)d_in[19];
  const float* We2 = (const float*)d_in[20];
  const float* Ws2 = (const float*)d_in[21], *bs2 = (const float*)d_in[22];
  const float* Wfp = (const float*)d_in[23], *bfp = (const float*)d_in[24];
  const float* Wfin= (const float*)d_in[25], *bfin= (const float*)d_in[26];

  const int* src = eidx;
  const int* dst = eidx + E;

  // Workspace layout (floats). Layer-2 q/k/v/s reuse the layer-1 region.
  float* ws = (float*)d_ws;
  size_t off = 0;
  float* bigA  = ws + off; off += (size_t)4 * N * 1024 + (size_t)N * 1024; // qkvs2+acc2
  float* h1    = ws + off; off += (size_t)N * 512;
  float* h2    = ws + off; off += (size_t)N * 1024;
  float* den   = ws + off; off += (size_t)N * 4;
  unsigned* amax = (unsigned*)(ws + off); off += (size_t)N * 4;
  float* alpha = ws + off; off += (size_t)E * 4;
  float* gcat  = ws + off; off += (size_t)Bg * 1280;

  float* q1 = bigA;                  // N x 512 each
  float* k1 = bigA + (size_t)N*512;
  float* v1 = bigA + (size_t)2*N*512;
  float* s1 = bigA + (size_t)3*N*512;
  float* acc1 = bigA + (size_t)4*N*512;
  float* q2 = bigA;                  // N x 1024 each (after h1 is formed)
  float* k2 = bigA + (size_t)N*1024;
  float* v2 = bigA + (size_t)2*N*1024;
  float* s2 = bigA + (size_t)3*N*1024;
  float* acc2 = bigA + (size_t)4*N*1024;

  const dim3 blk(GEMM_BLOCK);
  auto gemm_fast = [&](const float* A, int lda, const float* W, int ldw,
                       const float* bias, float* C, int ldc,
                       int M, int K, int Nn, int relu) {
    dim3 g((Nn + 255) / 256, M / 16);
    gemm_wmma_f32_fast<<<g, blk, 0, stream>>>(A, lda, W, ldw, bias, C, ldc, M, K, Nn, relu);
  };
  auto gemm_guard = [&](const float* A, int lda, const float* W, int ldw,
                        const float* bias, float* C, int ldc,
                        int M, int K, int Nn, int relu) {
    dim3 g((Nn + 255) / 256, M / 16);
    gemm_wmma_f32_guard<<<g, blk, 0, stream>>>(A, lda, W, ldw, bias, C, ldc, M, K, Nn, relu);
  };

  // ---- Layer 1: node projections (K=9 -> guarded/padded kernel) ----
  gemm_guard(x, 9, Wq1, 512, bq1, q1, 512, N, 9, 512, 0);
  gemm_guard(x, 9, Wk1, 512, bk1, k1, 512, N, 9, 512, 0);
  gemm_guard(x, 9, Wv1, 512, bv1, v1, 512, N, 9, 512, 0);
  gemm_guard(x, 9, Ws1, 512, bs1, s1, 512, N, 9, 512, 0);

  hipMemsetAsync(acc1, 0, (size_t)N * 512 * sizeof(float), stream);
  hipMemsetAsync(den,  0, (size_t)N * 4 * sizeof(float),   stream);
  hipMemsetAsync(amax, 0, (size_t)N * 4 * sizeof(unsigned), stream);

  const int edgeBlocks = (E * 4 + 7) / 8;   // 8 waves per block
  edge_alpha_max<<<edgeBlocks, 256, 0, stream>>>(src, dst, ea, We1, q1, k1,
                                                 alpha, amax, E, 4, 128,
                                                 0.08838834764831845f); // 1/sqrt(128)
  edge_scatter<<<edgeBlocks, 256, 0, stream>>>(src, dst, ea, We1, v1, alpha,
                                               amax, den, acc1, E, 4, 128);
  {
    const long tot = (long)N * 512;
    node_finalize<<<(unsigned)((tot + 255) / 256), 256, 0, stream>>>(
        acc1, den, s1, h1, tot, 512, 128, 4, /*relu=*/1);
  }

  // ---- Layer 2: node projections (K=512 -> fast async/WMMA kernel) ----
  gemm_fast(h1, 512, Wq2, 1024, bq2, q2, 1024, N, 512, 1024, 0);
  gemm_fast(h1, 512, Wk2, 1024, bk2, k2, 1024, N, 512, 1024, 0);
  gemm_fast(h1, 512, Wv2, 1024, bv2, v2, 1024, N, 512, 1024, 0);
  gemm_fast(h1, 512, Ws2, 1024, bs2, s2, 1024, N, 512, 1024, 0);

  hipMemsetAsync(acc2, 0, (size_t)N * 1024 * sizeof(float), stream);
  hipMemsetAsync(den,  0, (size_t)N * 4 * sizeof(float),    stream);
  hipMemsetAsync(amax, 0, (size_t)N * 4 * sizeof(unsigned), stream);

  edge_alpha_max<<<edgeBlocks, 256, 0, stream>>>(src, dst, ea, We2, q2, k2,
                                                 alpha, amax, E, 4, 256,
                                                 0.0625f); // 1/sqrt(256)
  edge_scatter<<<edgeBlocks, 256, 0, stream>>>(src, dst, ea, We2, v2, alpha,
                                               amax, den, acc2, E, 4, 256);
  {
    const long tot = (long)N * 1024;
    node_finalize<<<(unsigned)((tot + 255) / 256), 256, 0, stream>>>(
        acc2, den, s2, h2, tot, 1024, 256, 4, /*relu=*/0);
  }

  // ---- Pool + fingerprint + head ----
  pool_mean<<<(Bg * 1024 + 255) / 256, 256, 0, stream>>>(h2, gcat, Bg, NPG, 1024, 1280);
  gemm_fast(fp, 2048, Wfp, 256, bfp, gcat + 1024, 1280, Bg, 2048, 256, 0);
  gemm_fast(gcat, 1280, Wfin, 1024, bfin, (float*)d_out, 1024, Bg, 1280, 1024, 0);
}